// GWDLoss_29626684407920
// MI455X (gfx1250) — compile-verified
//
#include <hip/hip_runtime.h>
#include <hip/hip_bf16.h>
#include <stdint.h>

#define HW      65536
#define WIDTH   256
#define TOPK    100
#define NT      1024
#define NCAND   256
#define BUFCAP  2048
#define NCHUNK  8
#define CROWS   (WIDTH / NCHUNK)       // 32 rows per chunk
#define CELEMS  (HW / NCHUNK)          // 8192 floats = 32 KB per chunk

#if defined(__HIP_DEVICE_COMPILE__)
#  if __has_builtin(__builtin_amdgcn_tensor_load_to_lds)
#    define HAVE_TDM 1
#  endif
#  if __has_builtin(__builtin_amdgcn_s_wait_tensorcnt)
#    define HAVE_TCNT 1
#  endif
#endif

typedef unsigned int u32x4 __attribute__((ext_vector_type(4)));
typedef int          i32x4v __attribute__((ext_vector_type(4)));
typedef int          i32x8v __attribute__((ext_vector_type(8)));

__device__ __forceinline__ unsigned fmap(float f) {
    unsigned u = __float_as_uint(f);
    return (u & 0x80000000u) ? ~u : (u | 0x80000000u);
}

// Wait until TENSORcnt <= n (immediate required by the instruction encoding).
__device__ __forceinline__ void wait_tensorcnt_n(int n) {
#if defined(HAVE_TCNT)
    switch (n) {
        case 0: __builtin_amdgcn_s_wait_tensorcnt(0); break;
        case 1: __builtin_amdgcn_s_wait_tensorcnt(1); break;
        case 2: __builtin_amdgcn_s_wait_tensorcnt(2); break;
        case 3: __builtin_amdgcn_s_wait_tensorcnt(3); break;
        case 4: __builtin_amdgcn_s_wait_tensorcnt(4); break;
        case 5: __builtin_amdgcn_s_wait_tensorcnt(5); break;
        case 6: __builtin_amdgcn_s_wait_tensorcnt(6); break;
        default: __builtin_amdgcn_s_wait_tensorcnt(7); break;
    }
#else
    switch (n) {
        case 0: asm volatile("s_wait_tensorcnt 0" ::: "memory"); break;
        case 1: asm volatile("s_wait_tensorcnt 1" ::: "memory"); break;
        case 2: asm volatile("s_wait_tensorcnt 2" ::: "memory"); break;
        case 3: asm volatile("s_wait_tensorcnt 3" ::: "memory"); break;
        case 4: asm volatile("s_wait_tensorcnt 4" ::: "memory"); break;
        case 5: asm volatile("s_wait_tensorcnt 5" ::: "memory"); break;
        case 6: asm volatile("s_wait_tensorcnt 6" ::: "memory"); break;
        default: asm volatile("s_wait_tensorcnt 7" ::: "memory"); break;
    }
#endif
}

// One workgroup per (batch, which-heatmap). Streams the 256KB tile into LDS
// via 8 pipelined Tensor-Data-Mover descriptors (in-order completion), fusing
// the top-byte histogram scan with the transfer. Then one vectorized
// compaction scan, 3 radix passes over the tiny compacted buffer, and a
// bitonic-256 sort reproducing jax.top_k ordering/tie-breaking.
__global__ __launch_bounds__(NT, 1) void topk_kernel(const float* __restrict__ hm_p,
                                                     const float* __restrict__ hm_t,
                                                     float* __restrict__ coords) {
    extern __shared__ float tile[];                 // HW floats = 256 KB (dynamic)
    __shared__ unsigned hist[256];
    __shared__ unsigned long long buf[BUFCAP];      // compacted (u,idx) keys
    __shared__ unsigned long long cand[NCAND];
    __shared__ unsigned nbuf, cand_cnt;
    __shared__ unsigned s_prefix, s_pmask, s_krem;

    const int b     = blockIdx.x;                   // 0..63
    const int which = blockIdx.y;                   // 0 = hm_p, 1 = hm_t
    const float* src = (which == 0 ? hm_p : hm_t) + (size_t)b * HW;
    const int tid = threadIdx.x;

    // ---- Issue 8 chunked TDM loads (wave 0 only; TENSOR ops ignore EXEC) --
#if defined(HAVE_TDM)
    if (tid < 32) {
        uint64_t ga   = (uint64_t)(const void*)src;
        unsigned ldsb = (unsigned)(size_t)(void*)tile;    // low 32 bits = LDS byte offset
        i32x8v g1;
        g1[0] = 0x00020000;                   // data_size=4B, no flags
        g1[1] = (int)(WIDTH << 16);           // tensor_dim0 = 256
        g1[2] = (int)(CROWS << 16);           // tensor_dim1 = 32
        g1[3] = (int)(WIDTH << 16);           // tile_dim0   = 256
        g1[4] = (int)CROWS;                   // tile_dim1   = 32
        g1[5] = (int)WIDTH;                   // tensor_dim0_stride = 256
        g1[6] = 0;
        g1[7] = 0;
        i32x4v z4 = {0, 0, 0, 0};
#if __clang_major__ >= 23
        i32x8v z8 = {0, 0, 0, 0, 0, 0, 0, 0};
#endif
        for (int c = 0; c < NCHUNK; ++c) {
            uint64_t gac = ga + (uint64_t)c * (CELEMS * 4);
            u32x4 g0;
            g0[0] = 1u;                                   // count=1, user descriptor
            g0[1] = ldsb + (unsigned)c * (CELEMS * 4);    // lds_addr
            g0[2] = (unsigned)gac;                        // global_addr[31:0]
            g0[3] = ((unsigned)(gac >> 32) & 0x01FFFFFFu) // global_addr[56:32]
                  | 0x80000000u;                          // type=2 ("image")
#if __clang_major__ >= 23
            __builtin_amdgcn_tensor_load_to_lds(g0, g1, z4, z4, z8, 0);
#else
            __builtin_amdgcn_tensor_load_to_lds(g0, g1, z4, z4, 0);
#endif
        }
    }
#else
    for (int i = tid; i < HW; i += NT) tile[i] = src[i];
#endif

    if (tid == 0) { s_prefix = 0; s_pmask = 0; s_krem = TOPK; nbuf = 0; cand_cnt = 0; }
    if (tid < 256) hist[tid] = 0;
    __syncthreads();

    // ---- Scan 1: top-byte histogram, pipelined against the TDM stream -----
#if defined(HAVE_TDM)
    for (int c = 0; c < NCHUNK; ++c) {
        if (tid < 32) wait_tensorcnt_n(NCHUNK - 1 - c);   // chunks complete in order
        __syncthreads();
        const float4* c4 = (const float4*)(tile + c * CELEMS);
        for (int i = tid; i < CELEMS / 4; i += NT) {
            float4 v = c4[i];
            atomicAdd(&hist[fmap(v.x) >> 24], 1u);
            atomicAdd(&hist[fmap(v.y) >> 24], 1u);
            atomicAdd(&hist[fmap(v.z) >> 24], 1u);
            atomicAdd(&hist[fmap(v.w) >> 24], 1u);
        }
    }
#else
    {
        const float4* t4f = (const float4*)tile;
        for (int i = tid; i < HW / 4; i += NT) {
            float4 v = t4f[i];
            atomicAdd(&hist[fmap(v.x) >> 24], 1u);
            atomicAdd(&hist[fmap(v.y) >> 24], 1u);
            atomicAdd(&hist[fmap(v.z) >> 24], 1u);
            atomicAdd(&hist[fmap(v.w) >> 24], 1u);
        }
    }
#endif
    __syncthreads();
    if (tid == 0) {
        unsigned cum = 0;
        for (int bi = 255; bi >= 0; --bi) {
            unsigned c = hist[bi];
            if (cum + c >= s_krem) {
                s_prefix = ((unsigned)bi) << 24;
                s_pmask  = 0xFF000000u;
                s_krem  -= cum;
                break;
            }
            cum += c;
        }
    }
    __syncthreads();

    // ---- Scan 2: compact all elements with top byte >= b0 -----------------
    const unsigned thr0 = s_prefix;                  // b0 << 24
    const float4* t4 = (const float4*)tile;
    for (int i = tid; i < HW / 4; i += NT) {
        float4 v = t4[i];
        unsigned base = 4u * (unsigned)i;
        unsigned u0 = fmap(v.x), u1 = fmap(v.y), u2 = fmap(v.z), u3 = fmap(v.w);
        if (u0 >= thr0) { unsigned s = atomicAdd(&nbuf, 1u); if (s < BUFCAP) buf[s] = (((unsigned long long)u0) << 16) | (0xFFFFu ^ (base + 0)); }
        if (u1 >= thr0) { unsigned s = atomicAdd(&nbuf, 1u); if (s < BUFCAP) buf[s] = (((unsigned long long)u1) << 16) | (0xFFFFu ^ (base + 1)); }
        if (u2 >= thr0) { unsigned s = atomicAdd(&nbuf, 1u); if (s < BUFCAP) buf[s] = (((unsigned long long)u2) << 16) | (0xFFFFu ^ (base + 2)); }
        if (u3 >= thr0) { unsigned s = atomicAdd(&nbuf, 1u); if (s < BUFCAP) buf[s] = (((unsigned long long)u3) << 16) | (0xFFFFu ^ (base + 3)); }
    }
    __syncthreads();
    const unsigned n = (nbuf < BUFCAP) ? nbuf : BUFCAP;

    // ---- Radix passes 1..3 over the compacted buffer ----------------------
    for (int pass = 1; pass < 4; ++pass) {
        const int shift = 24 - 8 * pass;
        if (tid < 256) hist[tid] = 0;
        __syncthreads();
        const unsigned pref = s_prefix, pmask = s_pmask;
        for (unsigned j = tid; j < n; j += NT) {
            unsigned u = (unsigned)(buf[j] >> 16);
            if ((u & pmask) == pref) atomicAdd(&hist[(u >> shift) & 0xFFu], 1u);
        }
        __syncthreads();
        if (tid == 0) {
            unsigned cum = 0;
            for (int bi = 255; bi >= 0; --bi) {
                unsigned c = hist[bi];
                if (cum + c >= s_krem) {
                    s_prefix |= ((unsigned)bi) << shift;
                    s_pmask  |= 0xFFu << shift;
                    s_krem   -= cum;
                    break;
                }
                cum += c;
            }
        }
        __syncthreads();
    }

    // ---- Collect final candidates (u >= exact 100th-largest value) --------
    const unsigned T = s_prefix;
    for (unsigned j = tid; j < n; j += NT) {
        unsigned long long key = buf[j];
        if ((unsigned)(key >> 16) >= T) {
            unsigned slot = atomicAdd(&cand_cnt, 1u);
            if (slot < NCAND) cand[slot] = key;
        }
    }
    __syncthreads();
    {
        unsigned m = cand_cnt;
        if (tid < NCAND && (unsigned)tid >= m) cand[tid] = 0ULL;  // pad with minimum
    }

    // ---- Bitonic sort NCAND keys descending ((u desc, index asc)) ---------
    for (unsigned k = 2; k <= NCAND; k <<= 1) {
        for (unsigned j = k >> 1; j > 0; j >>= 1) {
            __syncthreads();
            if (tid < NCAND) {
                unsigned l = tid ^ j;
                if (l > (unsigned)tid) {
                    unsigned long long a = cand[tid], c = cand[l];
                    bool desc = ((tid & k) == 0);
                    if (desc ? (a < c) : (a > c)) { cand[tid] = c; cand[l] = a; }
                }
            }
        }
    }
    __syncthreads();

    // ---- Emit x/y of top-100 in rank order --------------------------------
    if (tid < TOPK) {
        unsigned idx = 0xFFFFu ^ (unsigned)(cand[tid] & 0xFFFFu);
        float* ob = coords + ((size_t)which * 64 + b) * 200;
        ob[tid]        = (float)(idx % WIDTH);   // xs
        ob[100 + tid]  = (float)(idx / WIDTH);   // ys
    }
}

__device__ __forceinline__ float gwd_one(float px, float py, float pw, float ph, float pr,
                                         float tx, float ty, float tw, float th, float tr) {
    const float DEG = 0.017453292519943295f;
    pw = fminf(fmaxf(pw, 1e-7f), 1e7f);  ph = fminf(fmaxf(ph, 1e-7f), 1e7f);
    tw = fminf(fmaxf(tw, 1e-7f), 1e7f);  th = fminf(fmaxf(th, 1e-7f), 1e7f);
    float ap = 0.5f * pw, bp = 0.5f * ph, at = 0.5f * tw, bt = 0.5f * th;
    float rp = pr * DEG, rt = tr * DEG;
    float cp = __cosf(rp), sp = __sinf(rp), ct = __cosf(rt), st = __sinf(rt);
    float ap2 = ap * ap, bp2 = bp * bp, at2 = at * at, bt2 = bt * bt;
    float Sp00 = ap2 * cp * cp + bp2 * sp * sp;
    float Sp11 = ap2 * sp * sp + bp2 * cp * cp;
    float Sp01 = (ap2 - bp2) * cp * sp;
    float St00 = at2 * ct * ct + bt2 * st * st;
    float St11 = at2 * st * st + bt2 * ct * ct;
    float St01 = (at2 - bt2) * ct * st;
    float trPT = Sp00 * St00 + 2.0f * Sp01 * St01 + Sp11 * St11;
    float whr  = ap2 + bp2 + at2 + bt2
               - 2.0f * sqrtf(fmaxf(trPT + 2.0f * (ap * bp * at * bt), 0.0f));
    float dx = px - tx, dy = py - ty;
    float dist = fmaxf(dx * dx + dy * dy + whr, 0.0f);
    float d = log1pf(dist);
    return 1.0f - 1.0f / (1.0f + d);
}

__global__ void loss_kernel(const float* __restrict__ ab_p, const float* __restrict__ ang_p,
                            const float* __restrict__ ab_t, const float* __restrict__ ang_t,
                            const int* __restrict__ ind, const int* __restrict__ reg_mask,
                            const float* __restrict__ coords, float* __restrict__ acc) {
    __shared__ float rl[256], rm[256];
    int g = blockIdx.x * blockDim.x + threadIdx.x;   // 0..6399
    float lsum = 0.0f, msum = 0.0f;
    if (g < 64 * TOPK) {
        int b = g / TOPK, k = g % TOPK;
        float m = (float)reg_mask[g];
        int id = ind[g];
        float xp = coords[(size_t)b * 200 + k];
        float yp = coords[(size_t)b * 200 + 100 + k];
        float xt = coords[((size_t)64 + b) * 200 + k];
        float yt = coords[((size_t)64 + b) * 200 + 100 + k];
        size_t hb = (size_t)b * 2 * HW;
        float pab0 = ab_p[hb + id];
        float pab1 = ab_p[hb + HW + id];
        float pang = ang_p[(size_t)b * HW + id];
        float tab0 = ab_t[(size_t)g * 2 + 0];
        float tab1 = ab_t[(size_t)g * 2 + 1];
        float tang = ang_t[g];
        lsum = gwd_one(xp * m, yp * m, pab0 * 2.0f * m, pab1 * 2.0f * m, (pang - 90.0f) * m,
                       xt * m, yt * m, tab0 * 2.0f * m, tab1 * 2.0f * m, (tang - 90.0f) * m);
        msum = m;
    }
    rl[threadIdx.x] = lsum;
    rm[threadIdx.x] = msum;
    __syncthreads();
    for (int s = 128; s > 0; s >>= 1) {
        if ((int)threadIdx.x < s) {
            rl[threadIdx.x] += rl[threadIdx.x + s];
            rm[threadIdx.x] += rm[threadIdx.x + s];
        }
        __syncthreads();
    }
    if (threadIdx.x == 0) {
        atomicAdd(&acc[0], rl[0]);
        atomicAdd(&acc[1], rm[0]);
    }
}

__global__ void init_kernel(float* acc) { acc[0] = 0.0f; acc[1] = 0.0f; }

__global__ void finalize_kernel(const float* acc, float* out) {
    out[0] = acc[0] / (acc[1] + 1e-8f);
}

extern "C" void kernel_launch(void* const* d_in, const int* in_sizes, int n_in,
                              void* d_out, int out_size, void* d_ws, size_t ws_size,
                              hipStream_t stream) {
    const float* hm_p  = (const float*)d_in[0];
    const float* ab_p  = (const float*)d_in[1];
    const float* ang_p = (const float*)d_in[2];
    const float* hm_t  = (const float*)d_in[3];
    const float* ab_t  = (const float*)d_in[4];
    const float* ang_t = (const float*)d_in[5];
    const int*   ind   = (const int*)d_in[6];
    const int*   rmask = (const int*)d_in[7];
    float* out = (float*)d_out;

    float* wsf    = (float*)d_ws;
    float* acc    = wsf;          // 2 floats
    float* coords = wsf + 16;     // [2][64][2][100] = 25600 floats

    (void)hipFuncSetAttribute(reinterpret_cast<const void*>(&topk_kernel),
                              hipFuncAttributeMaxDynamicSharedMemorySize, HW * 4);

    init_kernel<<<1, 1, 0, stream>>>(acc);
    topk_kernel<<<dim3(64, 2), NT, HW * 4, stream>>>(hm_p, hm_t, coords);
    loss_kernel<<<25, 256, 0, stream>>>(ab_p, ang_p, ab_t, ang_t, ind, rmask, coords, acc);
    finalize_kernel<<<1, 1, 0, stream>>>(acc, out);
}